// RandomTimeWarp_69217692942482
// MI455X (gfx1250) — compile-verified
//
#include <hip/hip_runtime.h>

// Problem constants (match reference)
#define Bsz   256
#define Csz   16
#define Tsz   4096
#define NKNOT 6        // KNOT + 2
#define NSEG  5
#define DXK   819.0f   // uniform knot spacing = (T-1)/(n-1)

#ifndef __has_builtin
#define __has_builtin(x) 0
#endif

#if __has_builtin(__builtin_amdgcn_global_load_async_to_lds_b128)
#define USE_ASYNC_LDS 1
#else
#define USE_ASYNC_LDS 0
#endif

// ---------------------------------------------------------------------------
// Compile-time inverse of the not-a-knot spline system matrix (6x6, uniform
// spacing dx=819), double-precision Gauss-Jordan, stored as float constants.
// ---------------------------------------------------------------------------
struct MatF { float v[NKNOT][NKNOT]; };

constexpr MatF ainv_init() {
    double A[NKNOT][2 * NKNOT] = {};
    const double dx = 819.0;
    A[0][0] = dx;            // dx[1]
    A[0][1] = 2.0 * dx;      // xs[2]-xs[0]
    for (int j = 1; j < NKNOT - 1; ++j) {
        A[j][j - 1] = dx;
        A[j][j]     = 4.0 * dx;
        A[j][j + 1] = dx;
    }
    A[NKNOT - 1][NKNOT - 2] = 2.0 * dx;  // xs[-1]-xs[-3]
    A[NKNOT - 1][NKNOT - 1] = dx;        // dx[-2]
    for (int i = 0; i < NKNOT; ++i) A[i][NKNOT + i] = 1.0;

    for (int col = 0; col < NKNOT; ++col) {
        int piv = col;
        double best = A[col][col] < 0 ? -A[col][col] : A[col][col];
        for (int r = col + 1; r < NKNOT; ++r) {
            double m = A[r][col] < 0 ? -A[r][col] : A[r][col];
            if (m > best) { best = m; piv = r; }
        }
        if (piv != col)
            for (int k = 0; k < 2 * NKNOT; ++k) {
                double t = A[col][k]; A[col][k] = A[piv][k]; A[piv][k] = t;
            }
        double inv = 1.0 / A[col][col];
        for (int k = 0; k < 2 * NKNOT; ++k) A[col][k] *= inv;
        for (int r = 0; r < NKNOT; ++r) {
            if (r == col) continue;
            double f = A[r][col];
            if (f != 0.0)
                for (int k = 0; k < 2 * NKNOT; ++k) A[r][k] -= f * A[col][k];
        }
    }
    MatF out = {};
    for (int i = 0; i < NKNOT; ++i)
        for (int j = 0; j < NKNOT; ++j)
            out.v[i][j] = (float)A[i][NKNOT + j];
    return out;
}

constexpr MatF h_ainv = ainv_init();
__constant__ MatF c_ainv = h_ainv;

typedef float v2f __attribute__((ext_vector_type(2)));
typedef float v8f __attribute__((ext_vector_type(8)));

// ---------------------------------------------------------------------------
// Phase 1: per-(b,c) spline coefficients. Each wave32 handles 16 columns and
// performs s = A_inv(6x6) . b(6x16) as two V_WMMA_F32_16X16X4_F32 ops
// (16x16x8 padded). Output: 21 floats per (b,c): a3[5],a2[5],a1[5],a0[5],scale
// ---------------------------------------------------------------------------
__global__ __launch_bounds__(128) void warp_spline_coeffs(
        const float* __restrict__ warps,   // (B, 6, C)
        float* __restrict__ coef)          // (B*C, 24)
{
    const int lane = threadIdx.x & 31;
    const int wave = threadIdx.x >> 5;
    const int half = lane >> 4;          // 0: K lo pair, 1: K hi pair
    const int r16  = lane & 15;
    const int base = (blockIdx.x * (blockDim.x >> 5) + wave) * 16;
    const int col  = base + r16;         // (b*C + c) in [0, B*C)
    const int bb   = col >> 4;           // C == 16
    const int cc   = col & 15;

    // Per-lane RHS vector for its column (both wave halves compute same cols)
    const float* wp = warps + (bb * NKNOT) * Csz + cc;
    float y[NKNOT];
#pragma unroll
    for (int k = 0; k < NKNOT; ++k)
        y[k] = (DXK * (float)k) * wp[k * Csz];

    float sl[NSEG];
#pragma unroll
    for (int k = 0; k < NSEG; ++k)
        sl[k] = (y[k + 1] - y[k]) / DXK;

    float bv[NKNOT];
    // b0 = ((dx + 2*d0)*dx*sl0 + dx^2*sl1)/d0,  d0 = 2*dx
    bv[0] = ((DXK + 2.0f * (2.0f * DXK)) * DXK * sl[0] + (DXK * DXK) * sl[1])
            / (2.0f * DXK);
#pragma unroll
    for (int j = 1; j < NKNOT - 1; ++j)
        bv[j] = 3.0f * (DXK * sl[j - 1] + DXK * sl[j]);
    bv[NKNOT - 1] = ((DXK * DXK) * sl[NSEG - 2]
                     + (2.0f * (2.0f * DXK) + DXK) * DXK * sl[NSEG - 1])
                    / (2.0f * DXK);

    // A: 16x4 per op; lanes 0-15 -> K={0,1}/{4,5}; lanes 16-31 -> K={2,3}/{6,7}
    const int k0 = half * 2;
    v2f a0v, a1v, b0v, b1v;
    a0v.x = (r16 < NKNOT) ? c_ainv.v[r16][k0]     : 0.0f;
    a0v.y = (r16 < NKNOT) ? c_ainv.v[r16][k0 + 1] : 0.0f;
    b0v.x = bv[k0];
    b0v.y = bv[k0 + 1];
    a1v.x = (!half && r16 < NKNOT) ? c_ainv.v[r16][4] : 0.0f;
    a1v.y = (!half && r16 < NKNOT) ? c_ainv.v[r16][5] : 0.0f;
    b1v.x = half ? 0.0f : bv[4];
    b1v.y = half ? 0.0f : bv[5];

    v8f acc = {};
    acc = __builtin_amdgcn_wmma_f32_16x16x4_f32(false, a0v, false, b0v,
                                                (short)0, acc, false, false);
    acc = __builtin_amdgcn_wmma_f32_16x16x4_f32(false, a1v, false, b1v,
                                                (short)0, acc, false, false);

    // D layout: VGPR j, lanes 0-15 -> row M=j, col N=lane
    if (lane < 16) {
        float s[NKNOT];
#pragma unroll
        for (int j = 0; j < NKNOT; ++j) s[j] = acc[j];

        float a3[NSEG], a2[NSEG];
#pragma unroll
        for (int j = 0; j < NSEG; ++j) {
            float t = (s[j] + s[j + 1] - 2.0f * sl[j]) / DXK;
            a3[j] = t / DXK;
            a2[j] = (sl[j] - s[j]) / DXK - t;
        }
        // scale from tw(q = T-1): seg = NSEG-1, u = DXK
        float u   = DXK;
        float twl = ((a3[NSEG - 1] * u + a2[NSEG - 1]) * u + s[NSEG - 1]) * u
                    + y[NSEG - 1];
        float scale = ((float)(Tsz - 1)) / twl;

        float* o = coef + col * 24;
#pragma unroll
        for (int j = 0; j < NSEG; ++j) {
            o[j]      = a3[j];
            o[5 + j]  = a2[j];
            o[10 + j] = s[j];   // a1
            o[15 + j] = y[j];   // a0
        }
        o[20] = scale;
    }
}

// ---------------------------------------------------------------------------
// Phase 2: one block per (b,c) row. Stage x row via async global->LDS
// (ASYNCcnt path), build warp grid xp in LDS, then exact bisect_right +
// linear interp, float4 stores. Streaming 64MB in + 64MB out -> HBM-bound.
// ---------------------------------------------------------------------------
#if USE_ASYNC_LDS
typedef int v4i __attribute__((vector_size(4 * sizeof(int))));
typedef __attribute__((address_space(1))) v4i gv4i;   // global
typedef __attribute__((address_space(3))) v4i lv4i;   // LDS
#endif

__global__ __launch_bounds__(256) void warp_apply(
        const float* __restrict__ x,      // (B, C, T)
        const float* __restrict__ coef,   // (B*C, 24)
        float* __restrict__ out)          // (B, C, T)
{
    __shared__ __align__(16) float s_xp[Tsz];
    __shared__ __align__(16) float s_x[Tsz];
    __shared__ float s_cf[24];

    const int row = blockIdx.x;
    const int tid = threadIdx.x;

    if (tid < 21) s_cf[tid] = coef[row * 24 + tid];

#if USE_ASYNC_LDS
    {
        // global_load_async_to_lds_b128: 16B per lane per op, ASYNCcnt-tracked
        gv4i* gp = (gv4i*)(x + (size_t)row * Tsz);   // addrspacecast flat->AS1
        lv4i* lp = (lv4i*)(&s_x[0]);                 // addrspacecast flat->AS3
#pragma unroll
        for (int i = 0; i < Tsz / (4 * 256); ++i) {  // 4 iterations, v4i units
            int idx = tid + i * 256;
            __builtin_amdgcn_global_load_async_to_lds_b128(gp + idx, lp + idx,
                                                           0, 0);
        }
#if __has_builtin(__builtin_amdgcn_s_wait_asynccnt)
        __builtin_amdgcn_s_wait_asynccnt(0);
#else
        asm volatile("s_wait_asynccnt 0x0" ::: "memory");
#endif
    }
#else
    {
        const float4* xin = (const float4*)(x + (size_t)row * Tsz);
#pragma unroll
        for (int i = 0; i < Tsz / 4 / 256; ++i) {
            int idx = tid + i * 256;
            ((float4*)s_x)[idx] = xin[idx];
        }
    }
#endif
    __syncthreads();

    const float scale = s_cf[20];
    // compute warp grid xp into LDS, 4 consecutive q per thread, float4 DS ops
#pragma unroll
    for (int i = 0; i < Tsz / (256 * 4); ++i) {
        int qb = (i * 256 + tid) * 4;
        float4 r;
#pragma unroll
        for (int k = 0; k < 4; ++k) {
            int q   = qb + k;
            int seg = q / 819;
            if (seg > NSEG - 1) seg = NSEG - 1;
            float u  = (float)q - DXK * (float)seg;
            float tw = ((s_cf[seg] * u + s_cf[5 + seg]) * u + s_cf[10 + seg])
                           * u + s_cf[15 + seg];
            float xpv = scale * tw;
            (&r.x)[k] = fminf(fmaxf(xpv, 0.0f), (float)(Tsz - 1));
        }
        ((float4*)s_xp)[i * 256 + tid] = r;
    }
    __syncthreads();

    float* orow = out + (size_t)row * Tsz;
#pragma unroll
    for (int i = 0; i < Tsz / (256 * 4); ++i) {
        int qb = (i * 256 + tid) * 4;
        float4 r;
#pragma unroll
        for (int k = 0; k < 4; ++k) {
            int q   = qb + k;
            float v = (float)q;
            // bisect_right: 12 guarded steps cover interval of 4096 exactly
            int lo = 0, hi = Tsz;
#pragma unroll
            for (int it = 0; it < 12; ++it) {
                if (lo < hi) {
                    int mid = (lo + hi) >> 1;
                    if (v < s_xp[mid]) hi = mid; else lo = mid + 1;
                }
            }
            int j = lo - 1;
            j = j < 0 ? 0 : (j > Tsz - 2 ? Tsz - 2 : j);
            float x0  = s_xp[j];
            float den = s_xp[j + 1] - x0;
            float w   = den > 0.0f ? (v - x0) / den : 0.0f;
            float f0  = s_x[j];
            float f1  = s_x[j + 1];
            (&r.x)[k] = f0 + w * (f1 - f0);
        }
        ((float4*)orow)[i * 256 + tid] = r;
    }
}

extern "C" void kernel_launch(void* const* d_in, const int* in_sizes, int n_in,
                              void* d_out, int out_size, void* d_ws, size_t ws_size,
                              hipStream_t stream) {
    const float* x     = (const float*)d_in[0];   // (B, C, T) f32
    const float* warps = (const float*)d_in[1];   // (B, 6, C) f32
    float* out  = (float*)d_out;
    float* coef = (float*)d_ws;                   // needs (B*C)*24*4 = 384 KB

    // 4096 columns / 16 per wave = 256 waves; 4 waves per block -> 64 blocks
    warp_spline_coeffs<<<(Bsz * Csz) / (16 * 4), 128, 0, stream>>>(warps, coef);
    warp_apply<<<Bsz * Csz, 256, 0, stream>>>(x, coef, out);
}